// interactionModule_15015205666999
// MI455X (gfx1250) — compile-verified
//
#include <hip/hip_runtime.h>
#include <math.h>

// ---------------------------------------------------------------------------
// InteractionModule graph op for MI455X (gfx1250, wave32).
//   per edge e: dt = x[src[e]] - theta[dst[e]];  accumulate (cos dt, sin dt)
//   into per-dst-node sums (count cancels under L2 normalization).
//   per node i: out = [cos(theta), sin(theta), sum_y / max(||sum||, eps)]
// Edge index streaming uses the CDNA5 async global->LDS DMA path
// (global_load_async_to_lds_b128 + s_wait_asynccnt) with double buffering.
// ---------------------------------------------------------------------------

#define TILE        1024            // edges per tile
#define BLOCK       256             // threads per block (8 waves)
#define EPT         (TILE / BLOCK)  // 4 edges per thread per tile
#define EDGE_BLOCKS 1024

#if defined(__AMDGCN__) &&                                                    \
    __has_builtin(__builtin_amdgcn_global_load_async_to_lds_b128) &&          \
    __has_builtin(__builtin_amdgcn_s_wait_asynccnt)
#define USE_ASYNC 1
#else
#define USE_ASYNC 0
#endif

#if USE_ASYNC
// Builtin signature (from hipcc diagnostic): param 0 is
// 'int __vector(4) __device__ *'  == addrspace(1) pointer to 16B int vector,
// param 1 is the LDS-side (addrspace(3)) counterpart.
typedef int vint4 __attribute__((vector_size(16)));
typedef __attribute__((address_space(1))) vint4 g_vint4;
typedef __attribute__((address_space(3))) vint4 l_vint4;

__device__ __forceinline__ void async_ld16(const void* g, void* l) {
  // 16 bytes per lane: one wave instruction moves 512B global -> LDS.
  __builtin_amdgcn_global_load_async_to_lds_b128((g_vint4*)g, (l_vint4*)l, 0, 0);
}
#endif

// ---------------------------------------------------------------------------
// Edge kernel: segment-accumulate (cos, sin) messages into acc[2*dst + {0,1}]
// ---------------------------------------------------------------------------
__global__ void __launch_bounds__(BLOCK)
edge_kernel(const float* __restrict__ x, const float* __restrict__ theta,
            const int* __restrict__ esrc, const int* __restrict__ edst,
            float* __restrict__ acc, int nTiles) {
  const int tid = threadIdx.x;

#if USE_ASYNC
  __shared__ __align__(16) int sbuf[2][TILE];
  __shared__ __align__(16) int dbuf[2][TILE];

  int t = (int)blockIdx.x;
  if (t >= nTiles) return;

  // Prologue: DMA first tile into buffer 0 (2 async instrs per wave).
  async_ld16(esrc + (size_t)t * TILE + tid * 4, &sbuf[0][tid * 4]);
  async_ld16(edst + (size_t)t * TILE + tid * 4, &dbuf[0][tid * 4]);

  int cur = 0;
  for (; t < nTiles; t += (int)gridDim.x) {
    const int nxt = t + (int)gridDim.x;
    if (nxt < nTiles) {
      // Kick off next tile's DMA into the other buffer, then wait until only
      // those 2 new async ops remain outstanding -> current buffer is ready.
      async_ld16(esrc + (size_t)nxt * TILE + tid * 4, &sbuf[cur ^ 1][tid * 4]);
      async_ld16(edst + (size_t)nxt * TILE + tid * 4, &dbuf[cur ^ 1][tid * 4]);
      __builtin_amdgcn_s_wait_asynccnt(2);
    } else {
      __builtin_amdgcn_s_wait_asynccnt(0);
    }
    asm volatile("" ::: "memory");  // keep LDS reads below the wait

    // Each thread consumes exactly the 4 edges it DMA'd (lane-private LDS
    // range) -> no workgroup barrier needed, only the per-wave async counter.
#pragma unroll
    for (int k = 0; k < EPT; ++k) {
      const int s = sbuf[cur][tid * EPT + k];
      const int d = dbuf[cur][tid * EPT + k];
      const float dt = x[s] - theta[d];      // 4B gathers, L2-resident arrays
      float sn, cs;
      __sincosf(dt, &sn, &cs);               // v_sin_f32 / v_cos_f32
      atomicAdd(&acc[2 * (size_t)d + 0], cs);  // both atomics of an edge hit
      atomicAdd(&acc[2 * (size_t)d + 1], sn);  // the same cache line
    }
    cur ^= 1;
  }
#else
  for (int t = (int)blockIdx.x; t < nTiles; t += (int)gridDim.x) {
    const size_t base = (size_t)t * TILE + (size_t)tid * EPT;
    __builtin_prefetch(esrc + base + (size_t)gridDim.x * TILE, 0, 0);
    __builtin_prefetch(edst + base + (size_t)gridDim.x * TILE, 0, 0);
#pragma unroll
    for (int k = 0; k < EPT; ++k) {
      const int s = esrc[base + k];
      const int d = edst[base + k];
      const float dt = x[s] - theta[d];
      float sn, cs;
      __sincosf(dt, &sn, &cs);
      atomicAdd(&acc[2 * (size_t)d + 0], cs);
      atomicAdd(&acc[2 * (size_t)d + 1], sn);
    }
  }
#endif
}

// Tail edges (E % TILE), plain path.
__global__ void edge_tail_kernel(const float* __restrict__ x,
                                 const float* __restrict__ theta,
                                 const int* __restrict__ esrc,
                                 const int* __restrict__ edst,
                                 float* __restrict__ acc, int base, int count) {
  const int i = base + (int)(blockIdx.x * blockDim.x + threadIdx.x);
  if (i >= base + count) return;
  const int s = esrc[i];
  const int d = edst[i];
  const float dt = x[s] - theta[d];
  float sn, cs;
  __sincosf(dt, &sn, &cs);
  atomicAdd(&acc[2 * (size_t)d + 0], cs);
  atomicAdd(&acc[2 * (size_t)d + 1], sn);
}

// ---------------------------------------------------------------------------
// Node kernel: v = (cos th, sin th);  w = sum_y / max(||sum||, eps)
// (segment count cancels in the normalization; cnt==0 -> sum==0 -> w==0)
// ---------------------------------------------------------------------------
__global__ void node_kernel(const float* __restrict__ theta,
                            const float* __restrict__ acc,
                            float* __restrict__ out, int n) {
  const int i = (int)(blockIdx.x * blockDim.x + threadIdx.x);
  if (i >= n) return;
  float sn, cs;
  sincosf(theta[i], &sn, &cs);               // precise for the output v
  const float sx = acc[2 * (size_t)i + 0];
  const float sy = acc[2 * (size_t)i + 1];
  const float nrm = fmaxf(sqrtf(sx * sx + sy * sy), 1e-12f);
  out[3 * (size_t)i + 0] = cs;               // v0 * cos(theta)
  out[3 * (size_t)i + 1] = sn;               // v0 * sin(theta)
  out[3 * (size_t)i + 2] = sy / nrm;         // w0 * normalize(mean)[...,1]
}

// ---------------------------------------------------------------------------
extern "C" void kernel_launch(void* const* d_in, const int* in_sizes, int n_in,
                              void* d_out, int out_size, void* d_ws,
                              size_t ws_size, hipStream_t stream) {
  const float* x     = (const float*)d_in[0];
  const float* theta = (const float*)d_in[1];
  const int*   esrc  = (const int*)d_in[2];
  const int*   edst  = (const int*)d_in[3];
  const int N = in_sizes[0];
  const int E = in_sizes[2];

  float* acc = (float*)d_ws;        // [2*N] interleaved (sum_cos, sum_sin)
  float* out = (float*)d_out;       // [N, 3]

  (void)hipMemsetAsync(acc, 0, (size_t)2 * N * sizeof(float), stream);

  const int nTiles = E / TILE;
  const int rem    = E - nTiles * TILE;
  if (nTiles > 0) {
    const int blocks = nTiles < EDGE_BLOCKS ? nTiles : EDGE_BLOCKS;
    edge_kernel<<<blocks, BLOCK, 0, stream>>>(x, theta, esrc, edst, acc, nTiles);
  }
  if (rem > 0) {
    edge_tail_kernel<<<(rem + BLOCK - 1) / BLOCK, BLOCK, 0, stream>>>(
        x, theta, esrc, edst, acc, nTiles * TILE, rem);
  }
  node_kernel<<<(N + BLOCK - 1) / BLOCK, BLOCK, 0, stream>>>(theta, acc, out, N);
}